// GCN_747324309853
// MI455X (gfx1250) — compile-verified
//
#include <hip/hip_runtime.h>

#define D 128

typedef float v2f __attribute__((ext_vector_type(2)));
typedef float v8f __attribute__((ext_vector_type(8)));

__device__ inline v8f wmma_f32_16x16x4(v2f a, v2f b, v8f c) {
    // (neg_a, A, neg_b, B, c_mod, C, reuse_a, reuse_b)
    return __builtin_amdgcn_wmma_f32_16x16x4_f32(false, a, false, b, (short)0, c, false, false);
}

__device__ inline void zero8(v8f& v) {
#pragma unroll
    for (int j = 0; j < 8; ++j) v[j] = 0.0f;
}

// LDS weight layout: pair-interleaved so one B fragment (rows 2p,2p+1 at column c)
// is a contiguous, 8B-aligned v2f:  Wl[p*256 + c*2 + j] = B[2p + j][c]
__device__ inline void fill_lds_rowmajor(float* Wl, const float* __restrict__ src, int tid) {
    for (int i = tid; i < D * D; i += 256) {
        int k = i >> 7, c = i & 127;
        Wl[(k >> 1) * 256 + c * 2 + (k & 1)] = src[i];  // B[k][c] = src[k*D+c]
    }
}
__device__ inline void fill_lds_transposed(float* Wl, const float* __restrict__ src, int tid) {
    for (int i = tid; i < D * D; i += 256) {
        int o = i >> 7, k = i & 127;
        Wl[(k >> 1) * 256 + o * 2 + (k & 1)] = src[i];  // B[k][o] = src[o*D+k]
    }
}

// 16x128 row-tile GEMM: acc[8] (16x16 tiles along N) += A(16x128) * B(128x128 in LDS).
// A fragment double-buffered to overlap global load latency with WMMA+DS work.
__device__ inline void gemm_tile(const float* __restrict__ arow, const float* __restrict__ Wl,
                                 int koff, int col, v8f acc[8]) {
    v2f a_cur;
    a_cur.x = arow[koff];
    a_cur.y = arow[koff + 1];
    for (int k = 0; k < D; k += 4) {
        const int kn = (k + 4 < D) ? (k + 4) : 0;  // last prefetch redundant, never used
        v2f a_next;
        a_next.x = arow[kn + koff];
        a_next.y = arow[kn + koff + 1];
        const int prow = (k + koff) >> 1;
#pragma unroll
        for (int t = 0; t < 8; ++t) {
            v2f b = *(const v2f*)&Wl[prow * 256 + (t * 16 + col) * 2];  // ds_load_b64
            acc[t] = wmma_f32_16x16x4(a_cur, b, acc[t]);
        }
        a_cur = a_next;
    }
}

// ---- K1: W[r] = sum_b comp[r,b] * basis[b]  ([R,D,D]) ----
__global__ void k_makeW(const float* __restrict__ basis, const float* __restrict__ comp,
                        float* __restrict__ W, int R, int B) {
    int i = blockIdx.x * 256 + threadIdx.x;
    if (i >= R * D * D) return;
    int r = i / (D * D);
    int io = i % (D * D);
    float s = 0.0f;
    for (int b = 0; b < B; ++b) s += comp[r * B + b] * basis[(size_t)b * D * D + io];
    W[i] = s;
}

// ---- K2: per-(dst,rel) counts + per-relation bucket counts ----
__global__ void k_hist(const int* __restrict__ dst, const int* __restrict__ et,
                       unsigned* __restrict__ cnt, unsigned* __restrict__ bc, int E, int R) {
    int e = blockIdx.x * 256 + threadIdx.x;
    if (e >= E) return;
    int r = et[e];
    atomicAdd(&cnt[(size_t)dst[e] * R + r], 1u);
    atomicAdd(&bc[r], 1u);
}

// ---- K3: exclusive prefix over R buckets (tiny) ----
__global__ void k_prefix(const unsigned* __restrict__ bc, unsigned* __restrict__ bbase, int R) {
    if (threadIdx.x == 0) {
        unsigned acc = 0;
        for (int r = 0; r < R; ++r) { bbase[r] = acc; acc += bc[r]; }
    }
}

// ---- K4: scatter edge ids into relation buckets ----
__global__ void k_scatter(const int* __restrict__ et, const unsigned* __restrict__ bbase,
                          unsigned* __restrict__ bfill, int* __restrict__ perm, int E) {
    int e = blockIdx.x * 256 + threadIdx.x;
    if (e >= E) return;
    int r = et[e];
    unsigned pos = atomicAdd(&bfill[r], 1u);
    perm[bbase[r] + pos] = e;
}

// ---- K5: fused RGCN message GEMM + mean-normalized scatter ----
// grid.x tiles of 128 edges, grid.y = relation. 8 waves/block, 16 edges/wave.
__global__ __launch_bounds__(256) void k_rgcn_msg(
    const float* __restrict__ x, const int* __restrict__ src, const int* __restrict__ dst,
    const float* __restrict__ Wall, const int* __restrict__ perm,
    const unsigned* __restrict__ cnt, const unsigned* __restrict__ bucket_cnt,
    const unsigned* __restrict__ bucket_base, float* __restrict__ agg, int R) {
    __shared__ float Wl[D * D];
    const int r = blockIdx.y;
    const unsigned bcnt = bucket_cnt[r];
    const unsigned blockStart = blockIdx.x * 128u;
    if (blockStart >= bcnt) return;  // block-uniform
    const unsigned bbase = bucket_base[r];

    fill_lds_rowmajor(Wl, Wall + (size_t)r * D * D, threadIdx.x);
    __syncthreads();

    const int wave = threadIdx.x >> 5;
    const int lane = threadIdx.x & 31;
    const unsigned rowStart = blockStart + (unsigned)(wave * 16);
    if (rowStart >= bcnt) return;  // wave-uniform; no barriers after this point

    // lanes 0..15 fetch edge meta for the wave's 16 edge-rows
    const int m = lane & 15;
    int e = -1, sN = 0, dN = 0;
    float nrm = 0.0f;
    const unsigned ge = rowStart + (unsigned)m;
    if (lane < 16 && ge < bcnt) {
        e = perm[bbase + ge];
        sN = src[e];
        dN = dst[e];
        unsigned c = cnt[(size_t)dN * R + r];
        nrm = 1.0f / (float)(c > 0u ? c : 1u);
    }
    const int srcRow = __shfl(sN, m, 32);  // A-row this lane feeds
    const int koff = (lane >> 4) * 2;      // lanes 0-15: K{0,1}; 16-31: K{2,3}
    const int col = lane & 15;

    v8f acc[8];
#pragma unroll
    for (int t = 0; t < 8; ++t) zero8(acc[t]);

    gemm_tile(x + (size_t)srcRow * D, Wl, koff, col, acc);

    // epilogue: scale each edge-row by its mean-norm, atomic-add into agg[dst]
#pragma unroll
    for (int v = 0; v < 8; ++v) {
        const int mrow = v + ((lane >> 4) << 3);
        const float nm = __shfl(nrm, mrow, 32);
        const int dd = __shfl(dN, mrow, 32);
        const int ee = __shfl(e, mrow, 32);
        if (ee >= 0) {
#pragma unroll
            for (int t = 0; t < 8; ++t)
                atomicAdd(&agg[(size_t)dd * D + t * 16 + col], acc[t][v] * nm);
        }
    }
}

// ---- K6: h1 = x @ root + agg + bias1 ----
__global__ __launch_bounds__(256) void k_h1(
    const float* __restrict__ x, const float* __restrict__ root,
    const float* __restrict__ bias1, const float* __restrict__ agg,
    float* __restrict__ h1, int N) {
    __shared__ float Wl[D * D];
    fill_lds_rowmajor(Wl, root, threadIdx.x);
    __syncthreads();
    const int wave = threadIdx.x >> 5;
    const int lane = threadIdx.x & 31;
    const size_t rowBase = ((size_t)blockIdx.x * 8 + wave) * 16;
    if (rowBase >= (size_t)N) return;  // N % 16 == 0, wave-uniform
    const int m = lane & 15;
    const int koff = (lane >> 4) * 2;
    const int col = lane & 15;

    v8f acc[8];
#pragma unroll
    for (int t = 0; t < 8; ++t) zero8(acc[t]);

    gemm_tile(x + (rowBase + m) * D, Wl, koff, col, acc);

#pragma unroll
    for (int v = 0; v < 8; ++v) {
        const size_t row = rowBase + (size_t)(v + ((lane >> 4) << 3));
#pragma unroll
        for (int t = 0; t < 8; ++t) {
            const int c = t * 16 + col;
            h1[row * D + c] = acc[t][v] + agg[row * D + c] + bias1[c];
        }
    }
}

// ---- K7: nbr[dst] += h1[src] (add aggregation); 2 edges per block ----
__global__ __launch_bounds__(256) void k_nbr(const int* __restrict__ src,
                                             const int* __restrict__ dst,
                                             const float* __restrict__ h1,
                                             float* __restrict__ nbr, int E) {
    const int e = blockIdx.x * 2 + (threadIdx.x >> 7);
    if (e >= E) return;
    const int j = threadIdx.x & 127;
    atomicAdd(&nbr[(size_t)dst[e] * D + j], h1[(size_t)src[e] * D + j]);
}

// ---- K8: out = nbr @ w_rel^T + h1 @ w_root^T + b_rel ----
__global__ __launch_bounds__(256) void k_out(
    const float* __restrict__ nbr, const float* __restrict__ h1,
    const float* __restrict__ w_rel, const float* __restrict__ w_root,
    const float* __restrict__ b_rel, float* __restrict__ out, int N) {
    __shared__ float Wl[D * D];
    const int wave = threadIdx.x >> 5;
    const int lane = threadIdx.x & 31;
    const size_t rowBase = ((size_t)blockIdx.x * 8 + wave) * 16;
    const bool active = rowBase < (size_t)N;
    const int m = lane & 15;
    const int koff = (lane >> 4) * 2;
    const int col = lane & 15;

    v8f acc[8];
#pragma unroll
    for (int t = 0; t < 8; ++t) zero8(acc[t]);

    // phase 1: B = w_rel^T, A = nbr
    fill_lds_transposed(Wl, w_rel, threadIdx.x);
    __syncthreads();
    if (active) gemm_tile(nbr + (rowBase + m) * D, Wl, koff, col, acc);
    __syncthreads();

    // phase 2: B = w_root^T, A = h1, accumulate; epilogue adds b_rel
    fill_lds_transposed(Wl, w_root, threadIdx.x);
    __syncthreads();
    if (active) {
        gemm_tile(h1 + (rowBase + m) * D, Wl, koff, col, acc);
#pragma unroll
        for (int v = 0; v < 8; ++v) {
            const size_t row = rowBase + (size_t)(v + ((lane >> 4) << 3));
#pragma unroll
            for (int t = 0; t < 8; ++t) {
                const int c = t * 16 + col;
                out[row * D + c] = acc[t][v] + b_rel[c];
            }
        }
    }
}

extern "C" void kernel_launch(void* const* d_in, const int* in_sizes, int n_in,
                              void* d_out, int out_size, void* d_ws, size_t ws_size,
                              hipStream_t stream) {
    const float* x      = (const float*)d_in[0];
    const int*   eidx   = (const int*)d_in[1];
    const int*   etype  = (const int*)d_in[2];
    const float* basis  = (const float*)d_in[3];
    const float* comp   = (const float*)d_in[4];
    const float* root   = (const float*)d_in[5];
    const float* bias1  = (const float*)d_in[6];
    const float* w_rel  = (const float*)d_in[7];
    const float* b_rel  = (const float*)d_in[8];
    const float* w_root = (const float*)d_in[9];
    float* out = (float*)d_out;

    const int N  = in_sizes[0] / D;
    const int E  = in_sizes[2];
    const int Bn = in_sizes[3] / (D * D);
    const int R  = in_sizes[4] / Bn;
    const int* src = eidx;       // edge_index[0]
    const int* dst = eidx + E;   // edge_index[1]

    // workspace carve (256B aligned regions), ~56 MB total
    char* p = (char*)d_ws;
    auto carve = [&](size_t bytes) {
        char* q = p;
        p += (bytes + 255) & ~(size_t)255;
        return q;
    };
    float*    W     = (float*)carve((size_t)R * D * D * 4);
    unsigned* cnt   = (unsigned*)carve((size_t)N * R * 4);
    unsigned* bc    = (unsigned*)carve((size_t)R * 4);
    unsigned* bbase = (unsigned*)carve((size_t)R * 4);
    unsigned* bfill = (unsigned*)carve((size_t)R * 4);
    int*      perm  = (int*)carve((size_t)E * 4);
    float*    agg   = (float*)carve((size_t)N * D * 4);  // reused as nbr later
    float*    h1    = (float*)carve((size_t)N * D * 4);
    (void)ws_size; (void)n_in; (void)out_size;

    hipMemsetAsync(cnt,   0, (size_t)N * R * 4, stream);
    hipMemsetAsync(bc,    0, (size_t)R * 4, stream);
    hipMemsetAsync(bfill, 0, (size_t)R * 4, stream);
    hipMemsetAsync(agg,   0, (size_t)N * D * 4, stream);

    k_makeW  <<<(R * D * D + 255) / 256, 256, 0, stream>>>(basis, comp, W, R, Bn);
    k_hist   <<<(E + 255) / 256, 256, 0, stream>>>(dst, etype, cnt, bc, E, R);
    k_prefix <<<1, 32, 0, stream>>>(bc, bbase, R);
    k_scatter<<<(E + 255) / 256, 256, 0, stream>>>(etype, bbase, bfill, perm, E);

    dim3 g5((unsigned)((E + 127) / 128), (unsigned)R);
    k_rgcn_msg<<<g5, 256, 0, stream>>>(x, src, dst, W, perm, cnt, bc, bbase, agg, R);

    k_h1<<<(N + 127) / 128, 256, 0, stream>>>(x, root, bias1, agg, h1, N);

    // reuse agg as nbr accumulator (in-stream re-zero keeps graph deterministic order)
    float* nbr = agg;
    hipMemsetAsync(nbr, 0, (size_t)N * D * 4, stream);
    k_nbr<<<(E + 1) / 2, 256, 0, stream>>>(src, dst, h1, nbr, E);

    k_out<<<(N + 127) / 128, 256, 0, stream>>>(nbr, h1, w_rel, w_root, b_rel, out, N);
}